// DetectionLoss_23252952941269
// MI455X (gfx1250) — compile-verified
//
#include <hip/hip_runtime.h>
#include <cstdint>
#include <cstddef>

// Problem constants (fixed by the reference): NB=3 boxes, C=20 classes, S=7, D=35, B=8192.
#define NBOX 3
#define DVEC 35               // 5*NB + C
#define NBATCH 8192.0f
#define WAVES_PER_BLOCK 4     // 128 threads, wave32
#define CELLS_PER_WAVE 32

typedef float v2f __attribute__((ext_vector_type(2)));
typedef float v8f __attribute__((ext_vector_type(8)));
typedef int   v4i __attribute__((ext_vector_type(4)));
typedef __attribute__((address_space(1))) void as1_void;
typedef __attribute__((address_space(3))) void as3_void;
typedef __attribute__((address_space(1))) v4i  as1_v4i;
typedef __attribute__((address_space(3))) v4i  as3_v4i;

// --- CDNA5 feature probes (device pass only; host pass parses fallbacks) ---
#if defined(__HIP_DEVICE_COMPILE__) && __has_builtin(__builtin_amdgcn_global_load_async_to_lds_b128)
#define HAVE_ASYNC_LDS 1
#endif
#if defined(__HIP_DEVICE_COMPILE__) && __has_builtin(__builtin_amdgcn_s_wait_asynccnt)
#define HAVE_WAIT_ASYNC 1
#endif
#if defined(__HIP_DEVICE_COMPILE__) && __has_builtin(__builtin_amdgcn_wmma_f32_16x16x4_f32)
#define HAVE_WMMA_F32 1
#endif

// 16B global -> LDS copy: async CDNA5 path (ASYNCcnt) or plain vector copy fallback.
__device__ __forceinline__ void copy16_g2l(const float* __restrict__ g, float* l) {
#if defined(HAVE_ASYNC_LDS)
  __builtin_amdgcn_global_load_async_to_lds_b128(
      (as1_v4i*)(as1_void*)g, (as3_v4i*)(as3_void*)l, 0, 0);
#else
  const float4 v = *(const float4*)g;
  *(float4*)l = v;
#endif
}

__device__ __forceinline__ void wait_g2l() {
#if defined(HAVE_ASYNC_LDS)
#if defined(HAVE_WAIT_ASYNC)
  __builtin_amdgcn_s_wait_asynccnt(0);
#else
  asm volatile("s_wait_asynccnt 0" ::: "memory");
#endif
#endif
  // non-async fallback: compiler tracks ds store->load deps itself
}

// Per-cell YOLO loss; o/t point at 35 contiguous floats (LDS or global).
__device__ __forceinline__ float cell_loss(const float* __restrict__ o,
                                           const float* __restrict__ t) {
  const float invS = 1.0f / 7.0f;
  const float conf_t = t[4];                       // t[4]==t[9]==t[14] in {0,1}
  const float obj   = (conf_t == 1.0f) ? 1.0f : 0.0f;
  const float noobj = (conf_t == 0.0f) ? 1.0f : 0.0f;

  // target box 0 -> xyxy
  const float tx = t[0] * invS, ty = t[1] * invS;
  const float thw = 0.5f * t[2], thh = 0.5f * t[3];
  const float tx0 = tx - thw, ty0 = ty - thh, tx1 = tx + thw, ty1 = ty + thh;
  const float a2 = (tx1 - tx0) * (ty1 - ty0);

  float best_iou = -1.0f;
  int best = 0;
  float noobj_sum = 0.0f;
#pragma unroll
  for (int b = 0; b < NBOX; ++b) {
    const float* p = o + 5 * b;
    const float* q = t + 5 * b;
    const float px = p[0] * invS, py = p[1] * invS;
    const float phw = 0.5f * p[2], phh = 0.5f * p[3];
    const float px0 = px - phw, py0 = py - phh, px1 = px + phw, py1 = py + phh;
    const float w = fmaxf(fminf(px1, tx1) - fmaxf(px0, tx0), 0.0f);
    const float h = fmaxf(fminf(py1, ty1) - fmaxf(py0, ty0), 0.0f);
    const float inter = w * h;
    const float a1 = (px1 - px0) * (py1 - py0);
    const float iou = inter / (a1 + a2 - inter);
    if (iou > best_iou) { best_iou = iou; best = b; }     // first-max like argmax
    const float dc = p[4] - q[4];
    noobj_sum += dc * dc;
  }
  const float* pb = o + 5 * best;
  const float* tb = t + 5 * best;
  const float dx = pb[0] - tb[0], dy = pb[1] - tb[1];
  const float xy = dx * dx + dy * dy;
  const float dw = sqrtf(pb[2]) - sqrtf(tb[2]);
  const float dh = sqrtf(pb[3]) - sqrtf(tb[3]);
  const float wh = dw * dw + dh * dh;
  const float dct = pb[4] - best_iou;
  float cls = 0.0f;
#pragma unroll
  for (int j = 5 * NBOX; j < DVEC; ++j) {
    const float d = o[j] - t[j];
    cls += d * d;
  }
  return obj * (5.0f * (xy + wh) + dct * dct + cls) + 0.5f * noobj * noobj_sum;
}

// Kernel 1: one cell per lane; wave stages its 32 cells (4480 B per tensor,
// 16B-aligned, fully coalesced) into LDS via CDNA5 async loads, computes, and
// block-reduces to one partial per block.
__global__ __launch_bounds__(WAVES_PER_BLOCK * 32) void yolo_loss_partial(
    const float* __restrict__ outp, const float* __restrict__ tgtp,
    int ncells, float* __restrict__ partial) {
  // 4 waves * 2 tensors * 32 cells * 35 floats = 35,840 B of LDS
  __shared__ __align__(16) float smem[WAVES_PER_BLOCK * 2 * CELLS_PER_WAVE * DVEC];
  __shared__ float red[WAVES_PER_BLOCK];

  const int tid = threadIdx.x;
  const int wave = tid >> 5;
  const int lane = tid & 31;
  const int cellBase = (blockIdx.x * WAVES_PER_BLOCK + wave) * CELLS_PER_WAVE;

  float* lout = &smem[wave * (2 * CELLS_PER_WAVE * DVEC)];
  float* ltgt = lout + CELLS_PER_WAVE * DVEC;

  float loss = 0.0f;
  if (cellBase + CELLS_PER_WAVE <= ncells) {           // uniform per wave
    const float* gout = outp + (size_t)cellBase * DVEC;
    const float* gtgt = tgtp + (size_t)cellBase * DVEC;
    // 32 cells * 35 floats = 1120 floats = 280 x 16B chunks per tensor
    for (int k = lane; k < 280; k += 32) {
      copy16_g2l(gout + 4 * k, lout + 4 * k);
      copy16_g2l(gtgt + 4 * k, ltgt + 4 * k);
    }
    wait_g2l();
    loss = cell_loss(lout + lane * DVEC, ltgt + lane * DVEC);
  } else {
    const int cell = cellBase + lane;
    if (cell < ncells)
      loss = cell_loss(outp + (size_t)cell * DVEC, tgtp + (size_t)cell * DVEC);
  }

#pragma unroll
  for (int m = 16; m >= 1; m >>= 1) loss += __shfl_xor(loss, m, 32);
  if (lane == 0) red[wave] = loss;
  __syncthreads();
  if (tid == 0) {
    float s = 0.0f;
#pragma unroll
    for (int w = 0; w < WAVES_PER_BLOCK; ++w) s += red[w];
    partial[blockIdx.x] = s;
  }
}

// Kernel 2: deterministic final reduction. Each wave reduces its 64 register
// partials with one v_wmma_f32_16x16x4_f32 (A = ones -> D rows = column sums).
__global__ __launch_bounds__(256) void yolo_loss_finalize(
    const float* __restrict__ partial, int np, float* __restrict__ result) {
  const int tid = threadIdx.x;
  float a0 = 0.0f, a1 = 0.0f;
  for (int i = tid; i < np; i += 512) a0 += partial[i];
  for (int i = tid + 256; i < np; i += 512) a1 += partial[i];

  float wsum;
#if defined(HAVE_WMMA_F32)
  v2f A = {1.0f, 1.0f};        // 16x4 all-ones
  v2f Bv = {a0, a1};           // 4x16 = this wave's 64 partials
  v8f C = {};
  // D[m][n] = sum_k B[k][n]; lane L, vgpr0 holds colsum(L%16)
  v8f D = __builtin_amdgcn_wmma_f32_16x16x4_f32(false, A, false, Bv, (short)0, C,
                                                false, false);
  wsum = D[0];
#pragma unroll
  for (int m = 16; m >= 1; m >>= 1) wsum += __shfl_xor(wsum, m, 32);
  wsum *= 0.5f;                // each colsum appears in 2 of 32 lanes
#else
  wsum = a0 + a1;
#pragma unroll
  for (int m = 16; m >= 1; m >>= 1) wsum += __shfl_xor(wsum, m, 32);
#endif

  __shared__ float red[8];
  if ((tid & 31) == 0) red[tid >> 5] = wsum;
  __syncthreads();
  if (tid == 0) {
    float tot = 0.0f;
#pragma unroll
    for (int w = 0; w < 8; ++w) tot += red[w];
    result[0] = tot * (1.0f / NBATCH);
  }
}

extern "C" void kernel_launch(void* const* d_in, const int* in_sizes, int n_in,
                              void* d_out, int out_size, void* d_ws, size_t ws_size,
                              hipStream_t stream) {
  const float* outp = (const float*)d_in[0];   // (B,S,S,D) f32
  const float* tgtp = (const float*)d_in[1];   // (B,S,S,D) f32
  float* result = (float*)d_out;               // 1 f32
  float* partial = (float*)d_ws;               // nblocks f32 scratch

  const int ncells = in_sizes[0] / DVEC;                       // 401,408
  const int nblocks = (ncells + (WAVES_PER_BLOCK * 32) - 1) /
                      (WAVES_PER_BLOCK * 32);                  // 3,136

  yolo_loss_partial<<<nblocks, WAVES_PER_BLOCK * 32, 0, stream>>>(outp, tgtp,
                                                                  ncells, partial);
  yolo_loss_finalize<<<1, 256, 0, stream>>>(partial, nblocks, result);
}